// GIN_44100724195438
// MI455X (gfx1250) — compile-verified
//
#include <hip/hip_runtime.h>
#include <math.h>

#define DIM 128
#define LDS_STRIDE 136   // 2*136 mod 64 == 16 -> half-waves hit disjoint bank ranges
#define BN_EPS 1e-5f
#define SCAN_T 1024

typedef float v2f __attribute__((ext_vector_type(2)));
typedef float v8f __attribute__((ext_vector_type(8)));

// ---------------------------------------------------------------- utilities
__global__ void k_zero(float* __restrict__ p, int n) {
  int i = blockIdx.x * blockDim.x + threadIdx.x;
  if (i < n) p[i] = 0.0f;
}

__global__ void k_izero(int* __restrict__ p, int n) {
  int i = blockIdx.x * blockDim.x + threadIdx.x;
  if (i < n) p[i] = 0;
}

__global__ void k_icopy(int* __restrict__ dst, const int* __restrict__ src, int n) {
  int i = blockIdx.x * blockDim.x + threadIdx.x;
  if (i < n) dst[i] = src[i];
}

// ------------------------------------------------------- CSR build (by dst)
__global__ void k_hist(int* __restrict__ deg, const int* __restrict__ edst, int E) {
  int e = blockIdx.x * blockDim.x + threadIdx.x;
  if (e < E) atomicAdd(&deg[edst[e]], 1);
}

// single-block exclusive scan of deg[0..M) -> rowptr[0..M], rowptr[M]=total
__global__ void __launch_bounds__(SCAN_T)
k_scan(const int* __restrict__ deg, int* __restrict__ rowptr, int M) {
  __shared__ int part[SCAN_T];
  int tid = threadIdx.x;
  int chunk = (M + SCAN_T - 1) / SCAN_T;
  int s0 = tid * chunk;
  int s1 = s0 + chunk; if (s1 > M) s1 = M;
  if (s0 > M) s0 = M;
  int s = 0;
  for (int i = s0; i < s1; ++i) s += deg[i];
  part[tid] = s;
  __syncthreads();
  for (int off = 1; off < SCAN_T; off <<= 1) {
    int v = (tid >= off) ? part[tid - off] : 0;
    __syncthreads();
    part[tid] += v;
    __syncthreads();
  }
  int run = (tid == 0) ? 0 : part[tid - 1];
  for (int i = s0; i < s1; ++i) {
    rowptr[i] = run;
    run += deg[i];
  }
  if (tid == SCAN_T - 1) rowptr[M] = run;
}

__global__ void k_fill(const int* __restrict__ esrc, const int* __restrict__ edst,
                       int* __restrict__ cursor, int* __restrict__ eidx, int E) {
  int e = blockIdx.x * blockDim.x + threadIdx.x;
  if (e >= E) return;
  int pos = atomicAdd(&cursor[edst[e]], 1);
  eidx[pos] = esrc[e];
}

// ------------------------------------------ GIN aggregate: a = h + gather-sum
// One block per node; 128 threads own one column each. Coalesced 512B row
// reads, register accumulation, zero float atomics.
__global__ void __launch_bounds__(128)
k_gather(float* __restrict__ a, const float* __restrict__ h,
         const int* __restrict__ rowptr, const int* __restrict__ eidx, int M) {
  int n = blockIdx.x;
  int d = threadIdx.x;
  float s = h[(size_t)n * DIM + d];
  int beg = rowptr[n], end = rowptr[n + 1];
  for (int i = beg; i < end; ++i) {
    int src = eidx[i];
    if ((i + 1 < end) && ((d & 31) == 0))  // global_prefetch_b8 next random row
      __builtin_prefetch(h + (size_t)eidx[i + 1] * DIM + d, 0, 1);
    s += h[(size_t)src * DIM + d];
  }
  a[(size_t)n * DIM + d] = s;
}

// ---------------------------------------------------------------- WMMA GEMM
// C[M x 128] = A[M x 128] @ W[128 x 128] + bias  (optional fused ReLU)
// V_WMMA_F32_16X16X4_F32: full fp32 (workload is HBM-bound, not FLOP-bound).
__global__ void __launch_bounds__(256)
k_gemm(const float* __restrict__ A, const float* __restrict__ W,
       const float* __restrict__ bias, float* __restrict__ C,
       int M, int relu) {
  extern __shared__ float Wlds[];  // DIM * LDS_STRIDE floats

  int tid = threadIdx.x;
  for (int i = tid; i < DIM * DIM; i += 256) {
    int r = i >> 7;
    int c = i & 127;
    Wlds[r * LDS_STRIDE + c] = W[i];
  }
  __syncthreads();

  int wave = tid >> 5;
  int lane = tid & 31;
  int strip = blockIdx.x * 8 + wave;
  int rowBase = strip * 16;

  int mn = lane & 15;        // A: row-in-tile, B/C: col-in-tile
  int kb = (lane >> 4) << 1; // K sub-offset 0 or 2 (ISA 16x4 f32 layout)

  int rowA = rowBase + mn;
  if (rowA >= M) rowA = M - 1;  // clamp loads for tail strips (stores guarded)
  const float* Arow = A + (size_t)rowA * DIM;

  v8f acc[8] = {};

  for (int kc = 0; kc < DIM; kc += 4) {
    v2f a = *(const v2f*)(Arow + kc + kb);
#pragma unroll
    for (int nt = 0; nt < 8; ++nt) {
      v2f b;
      b.x = Wlds[(kc + kb) * LDS_STRIDE + nt * 16 + mn];
      b.y = Wlds[(kc + kb + 1) * LDS_STRIDE + nt * 16 + mn];
      acc[nt] = __builtin_amdgcn_wmma_f32_16x16x4_f32(
          false, a, false, b, (short)0, acc[nt], false, false);
    }
  }

  if (rowBase < M) {
    int rOff = (lane >> 4) << 3;  // D layout: VGPR r -> row r (lanes 0-15) / r+8
#pragma unroll
    for (int nt = 0; nt < 8; ++nt) {
      int col = nt * 16 + mn;
      float bv = bias[col];
#pragma unroll
      for (int r = 0; r < 8; ++r) {
        float v = acc[nt][r] + bv;
        if (relu) v = fmaxf(v, 0.0f);
        C[(size_t)(rowBase + rOff + r) * DIM + col] = v;
      }
    }
  }
}

// ----------------------------------------------------- batch-norm (training)
__global__ void k_colstats(const float* __restrict__ T, float* __restrict__ sums,
                           float* __restrict__ sqs, int M) {
  int tid = threadIdx.x;
  int col = tid & 127;
  int base = blockIdx.x * 256;
  int end = base + 256;
  if (end > M) end = M;
  float s = 0.0f, q = 0.0f;
  for (int r = base + (tid >> 7); r < end; r += 2) {
    float v = T[(size_t)r * DIM + col];
    s += v;
    q += v * v;
  }
  atomicAdd(&sums[col], s);
  atomicAdd(&sqs[col], q);
}

__global__ void k_bnfinal(const float* __restrict__ sums, const float* __restrict__ sqs,
                          float* __restrict__ mean, float* __restrict__ istd, int M) {
  int i = threadIdx.x;
  float inv = 1.0f / (float)M;
  float m = sums[i] * inv;
  float v = sqs[i] * inv - m * m;
  mean[i] = m;
  istd[i] = rsqrtf(v + BN_EPS);
}

__global__ void k_bnrelu(float* __restrict__ out, const float* __restrict__ T,
                         const float* __restrict__ g, const float* __restrict__ b,
                         const float* __restrict__ mean, const float* __restrict__ istd,
                         int n) {
  int i = blockIdx.x * blockDim.x + threadIdx.x;
  if (i >= n) return;
  int c = i & 127;
  float v = g[c] * (T[i] - mean[c]) * istd[c] + b[c];
  out[i] = fmaxf(v, 0.0f);
}

// ------------------------------------------------- segment max pool (sorted)
__global__ void k_gbounds(const int* __restrict__ batch, int* __restrict__ gstart,
                          int M, int G) {
  int g = blockIdx.x * blockDim.x + threadIdx.x;
  if (g > G) return;
  if (g == G) { gstart[G] = M; return; }
  int lo = 0, hi = M;               // lower_bound(batch, g)
  while (lo < hi) {
    int mid = (lo + hi) >> 1;
    if (batch[mid] < g) lo = mid + 1; else hi = mid;
  }
  gstart[g] = lo;
}

__global__ void __launch_bounds__(128)
k_pool_seg(float* __restrict__ p, const float* __restrict__ h,
           const int* __restrict__ gstart, int G) {
  int g = blockIdx.x;
  int d = threadIdx.x;
  int beg = gstart[g], end = gstart[g + 1];
  float m = -INFINITY;
  for (int n = beg; n < end; ++n) m = fmaxf(m, h[(size_t)n * DIM + d]);
  p[(size_t)g * DIM + d] = m;
}

// --------------------------------------------------- final 128->2 projection
__global__ void k_rout(float* __restrict__ score, const float* __restrict__ q,
                       const float* __restrict__ wo, const float* __restrict__ bo,
                       int G) {
  int t = blockIdx.x * blockDim.x + threadIdx.x;
  if (t >= G * 2) return;
  int g = t >> 1;
  int o = t & 1;
  float s = bo[o];
  const float* row = q + (size_t)g * DIM;
#pragma unroll 8
  for (int k = 0; k < DIM; ++k) s += row[k] * wo[k * 2 + o];
  score[t] += s;
}

// ---------------------------------------------------------------- launchers
static inline void gemm_launch(const float* A, const float* W, const float* bias,
                               float* C, int M, int relu, hipStream_t stream) {
  int strips = (M + 15) / 16;
  int blocks = (strips + 7) / 8;
  size_t lds = (size_t)DIM * LDS_STRIDE * sizeof(float);
  k_gemm<<<blocks, 256, lds, stream>>>(A, W, bias, C, M, relu);
}

static inline void bn_relu(const float* T, float* out, const float* g, const float* b,
                           float* sums, float* sqs, float* mean, float* istd,
                           int M, hipStream_t stream) {
  k_zero<<<1, 256, 0, stream>>>(sums, 256);  // sums(128) + sqs(128) contiguous
  k_colstats<<<(M + 255) / 256, 256, 0, stream>>>(T, sums, sqs, M);
  k_bnfinal<<<1, 128, 0, stream>>>(sums, sqs, mean, istd, M);
  int n = M * DIM;
  k_bnrelu<<<(n + 255) / 256, 256, 0, stream>>>(out, T, g, b, mean, istd, n);
}

static inline void readout(const float* h, const int* gstart, int M, int G, int layer,
                           const float* r_w, const float* r_b, const float* r_wo,
                           const float* r_bo, float* P, float* Q1, float* Q2,
                           float* score, hipStream_t stream) {
  k_pool_seg<<<G, 128, 0, stream>>>(P, h, gstart, G);
  gemm_launch(P, r_w + ((size_t)layer * 2 + 0) * DIM * DIM,
              r_b + ((size_t)layer * 2 + 0) * DIM, Q1, G, 1, stream);
  gemm_launch(Q1, r_w + ((size_t)layer * 2 + 1) * DIM * DIM,
              r_b + ((size_t)layer * 2 + 1) * DIM, Q2, G, 1, stream);
  k_rout<<<(G * 2 + 255) / 256, 256, 0, stream>>>(
      score, Q2, r_wo + (size_t)layer * DIM * 2, r_bo + (size_t)layer * 2, G);
}

extern "C" void kernel_launch(void* const* d_in, const int* in_sizes, int n_in,
                              void* d_out, int out_size, void* d_ws, size_t ws_size,
                              hipStream_t stream) {
  (void)n_in; (void)ws_size;
  const float* x      = (const float*)d_in[0];
  const int*   ei     = (const int*)d_in[1];
  const int*   batch  = (const int*)d_in[2];
  const float* conv_w = (const float*)d_in[3];
  const float* conv_b = (const float*)d_in[4];
  const float* cbn_g  = (const float*)d_in[5];
  const float* cbn_b  = (const float*)d_in[6];
  const float* obn_g  = (const float*)d_in[7];
  const float* obn_b  = (const float*)d_in[8];
  const float* r_w    = (const float*)d_in[9];
  const float* r_b    = (const float*)d_in[10];
  const float* r_wo   = (const float*)d_in[11];
  const float* r_bo   = (const float*)d_in[12];

  const int M = in_sizes[0] / DIM;                  // 50000 (multiple of 16)
  const int E = in_sizes[1] / 2;                    // 800000
  const int G = out_size / 2;                       // 512 graphs
  const int NCONV = in_sizes[3] / (3 * DIM * DIM);  // 7

  float* ws = (float*)d_ws;
  size_t nd = (size_t)M * DIM;
  float* A    = ws;                 // aggregated / activations
  float* T    = A + nd;             // GEMM output (pre-BN)
  float* H    = T + nd;             // layer output
  float* P    = H + nd;             // pooled [G,128]
  float* Q1   = P + (size_t)G * DIM;
  float* Q2   = Q1 + (size_t)G * DIM;
  float* sums = Q2 + (size_t)G * DIM;
  float* sqs  = sums + DIM;
  float* mean = sqs + DIM;
  float* istd = mean + DIM;
  int* rowptr = (int*)(istd + DIM);   // M+1
  int* cursor = rowptr + (M + 1);     // M (doubles as deg histogram)
  int* eidx   = cursor + M;           // E (source node per CSR slot)
  int* gstart = eidx + E;             // G+1

  float* score = (float*)d_out;
  k_zero<<<(out_size + 255) / 256, 256, 0, stream>>>(score, out_size);

  const int* esrc = ei;
  const int* edst = ei + E;

  // ---- build CSR (by destination) once; reused by all 7 convs
  k_izero<<<(M + 255) / 256, 256, 0, stream>>>(cursor, M);        // deg = 0
  k_hist<<<(E + 255) / 256, 256, 0, stream>>>(cursor, edst, E);   // deg histogram
  k_scan<<<1, SCAN_T, 0, stream>>>(cursor, rowptr, M);            // exclusive scan
  k_icopy<<<(M + 255) / 256, 256, 0, stream>>>(cursor, rowptr, M);
  k_fill<<<(E + 255) / 256, 256, 0, stream>>>(esrc, edst, cursor, eidx, E);

  // ---- per-graph segment bounds (batch is sorted)
  k_gbounds<<<(G + 256) / 256, 256, 0, stream>>>(batch, gstart, M, G);

  // readout 0 on raw features
  readout(x, gstart, M, G, 0, r_w, r_b, r_wo, r_bo, P, Q1, Q2, score, stream);

  const float* hin = x;
  for (int i = 0; i < NCONV; ++i) {
    // a = h + sum_{j in N(n)} h[j]   (gather over CSR, no float atomics)
    k_gather<<<M, 128, 0, stream>>>(A, hin, rowptr, eidx, M);

    // MLP: (linear -> BN -> relu) x2
    for (int j = 0; j < 2; ++j) {
      const float* W = conv_w + ((size_t)i * 3 + j) * DIM * DIM;
      const float* B = conv_b + ((size_t)i * 3 + j) * DIM;
      gemm_launch(A, W, B, T, M, 0, stream);
      bn_relu(T, A, cbn_g + ((size_t)i * 2 + j) * DIM, cbn_b + ((size_t)i * 2 + j) * DIM,
              sums, sqs, mean, istd, M, stream);
    }
    // last linear, then outer BN + relu
    gemm_launch(A, conv_w + ((size_t)i * 3 + 2) * DIM * DIM,
                conv_b + ((size_t)i * 3 + 2) * DIM, T, M, 0, stream);
    bn_relu(T, H, obn_g + (size_t)i * DIM, obn_b + (size_t)i * DIM,
            sums, sqs, mean, istd, M, stream);

    hin = H;
    readout(H, gstart, M, G, i + 1, r_w, r_b, r_wo, r_bo, P, Q1, Q2, score, stream);
  }
}